// LocallyConnectedAutoencoder_24086176596711
// MI455X (gfx1250) — compile-verified
//
#include <hip/hip_runtime.h>
#include <hip/hip_bf16.h>

// LocallyConnectedAutoencoder fused kernel for gfx1250 (MI455X).
// Shapes: B=2048, H=256, W=128, P=32 -> TP=32 patches, PD=1024, HPP=32.
// encode: per patch GEMM [B x 1024] * [1024 x 32]   (K=1024, N=32)
// decode: per patch GEMM [B x 32]   * [32 x 1024]   (K=32,   N=1024)
// HBM-bound (AI ~16.5 F/B at 23.3 TB/s) => exact f32 math via
// V_WMMA_F32_16X16X4_F32, with double-buffered GLOBAL_LOAD_ASYNC_TO_LDS_B128
// staging so DMA overlaps the WMMA stream.

typedef float v2f __attribute__((ext_vector_type(2)));
typedef float v8f __attribute__((ext_vector_type(8)));
typedef int   v4i __attribute__((vector_size(16)));   // builtin uses GCC vec type

static __device__ __forceinline__ v8f wmma_f32(v2f a, v2f b, v8f c) {
  // 8 args: (neg_a, A, neg_b, B, c_mod, C, reuse_a, reuse_b)
  return __builtin_amdgcn_wmma_f32_16x16x4_f32(
      false, a, false, b, (short)0, c, false, false);
}

#if defined(__has_builtin)
#  if __has_builtin(__builtin_amdgcn_global_load_async_to_lds_b128)
#    define LCA_ASYNC 1
#  endif
#  if __has_builtin(__builtin_amdgcn_s_wait_asynccnt)
#    define LCA_HAVE_WAIT_BUILTIN 1
#  endif
#endif
#ifndef LCA_ASYNC
#  define LCA_ASYNC 0
#endif

#define AS1G __attribute__((address_space(1)))
#define AS3L __attribute__((address_space(3)))

#if LCA_ASYNC
// One wave32 instruction: each lane copies 16B memory -> LDS, tracked by ASYNCcnt.
static __device__ __forceinline__ void async_copy16(const float* g, float* l) {
  // Flat LDS address low 32 bits == LDS byte offset (ISA aperture rules).
  __builtin_amdgcn_global_load_async_to_lds_b128(
      (AS1G v4i*)(unsigned long long)g,
      (AS3L v4i*)(unsigned int)(unsigned long long)l,
      0, 0);
}
static __device__ __forceinline__ void wait_async0() {
#if defined(LCA_HAVE_WAIT_BUILTIN)
  __builtin_amdgcn_s_wait_asynccnt(0);
#else
  asm volatile("s_wait_asynccnt 0" ::: "memory");
#endif
}
#else
static __device__ __forceinline__ void async_copy16(const float* g, float* l) {
  *(float4*)l = *(const float4*)g;
}
static __device__ __forceinline__ void wait_async0() {}
#endif

#define HW_TOTAL 32768   // H*W = 256*128
#define T_STRIDE 32768   // per-patch weight block: 32*1024 or 1024*32
#define KC 64            // K-chunk for encode
#define XS_STRIDE 68     // 64 + 4 pad -> (4r+k) mod 64 conflict-free gathers
#define ES_STRIDE 36     // 32 + 4 pad
#define XS_FLOATS (64 * XS_STRIDE)            // 4352
#define WS_FLOATS (32 * XS_STRIDE)            // 2176
#define BUF_FLOATS (XS_FLOATS + WS_FLOATS)    // 6528
#define SMEM_FLOATS (2 * BUF_FLOATS)          // 13056 floats = 52224 B < 64 KB
#define ES_FLOATS (64 * ES_STRIDE)            // 2304

__global__ void __launch_bounds__(128)
lca_fused_kernel(const float* __restrict__ x,
                 const float* __restrict__ ew,   // [32][32][1024]
                 const float* __restrict__ eb,   // [32][32]
                 const float* __restrict__ dw,   // [32][1024][32]
                 const float* __restrict__ db,   // [32][1024]
                 float* __restrict__ out,
                 int Brows) {
  __shared__ float smem[SMEM_FLOATS];

  const int tid     = threadIdx.x;
  const int lane    = tid & 31;
  const int w       = tid >> 5;      // wave id 0..3
  const int l16     = lane & 15;
  const int halfSel = lane >> 4;     // 0 or 1

  const int bid = blockIdx.x;
  const int t   = bid & 31;          // patch id
  const int m   = bid >> 5;          // row-group id
  const int B0  = m * 64;

  const int ph    = t >> 2;          // t / NPW
  const int pw    = t & 3;           // t % NPW
  const int t_off = ph * 4096 + pw * 32;

  const float* xw_base = ew + (size_t)t * T_STRIDE;
  const float* dw_base = dw + (size_t)t * T_STRIDE;

  // Issue async stage of encode chunk c into buffer bsel.
  auto stage_enc = [&](int c, int bsel) {
    float* xs = smem + bsel * BUF_FLOATS;
    float* ws = xs + XS_FLOATS;
    // X tile: 64 rows x 64 k (= 2 image rows of 32) -> 1024 float4, 8/thread
    #pragma unroll
    for (int it = 0; it < 8; ++it) {
      int q  = tid + 128 * it;
      int jj = q & 7;            // float4 within a 32-float j-segment
      int il = (q >> 3) & 1;     // which image row of the chunk
      int r  = q >> 4;           // batch row 0..63
      const float* src = x + (size_t)(B0 + r) * HW_TOTAL + t_off
                           + (2 * c + il) * 128 + jj * 4;
      async_copy16(src, xs + r * XS_STRIDE + il * 32 + jj * 4);
    }
    // Encoder weight tile: 32 h x 64 k -> 512 float4, 4/thread
    #pragma unroll
    for (int it = 0; it < 4; ++it) {
      int q  = tid + 128 * it;
      int jj = q & 15;
      int h  = q >> 4;
      const float* src = xw_base + h * 1024 + c * KC + jj * 4;
      async_copy16(src, ws + h * XS_STRIDE + jj * 4);
    }
  };

  // ---------------- encode: C_enc[16x32] per wave, double-buffered ----------
  v8f c0 = {}; // N = h in [0,16)
  v8f c1 = {}; // N = h in [16,32)

  stage_enc(0, 0);
  wait_async0();
  __syncthreads();

  for (int c = 0; c < 16; ++c) {
    const int cur = c & 1;
    if (c + 1 < 16) stage_enc(c + 1, cur ^ 1);   // DMA next chunk during WMMAs

    const float* xs = smem + cur * BUF_FLOATS;
    const float* ws = xs + XS_FLOATS;
    const float* Xrow = xs + (w * 16 + l16) * XS_STRIDE;
    const float* Wn0  = ws + l16 * XS_STRIDE;
    const float* Wn1  = ws + (16 + l16) * XS_STRIDE;
    #pragma unroll
    for (int k2 = 0; k2 < KC; k2 += 4) {
      // A 16x4 frag: lanes 0-15 hold K=k2,k2+1 ; lanes 16-31 hold K=k2+2,k2+3
      int kA = k2 + 2 * halfSel;
      v2f a; a[0] = Xrow[kA]; a[1] = Xrow[kA + 1];
      // B 4x16 frag: VGPR0 = row k2+halfSel, VGPR1 = row k2+2+halfSel
      int kB = k2 + halfSel;
      v2f b0; b0[0] = Wn0[kB]; b0[1] = Wn0[kB + 2];
      v2f b1; b1[0] = Wn1[kB]; b1[1] = Wn1[kB + 2];
      c0 = wmma_f32(a, b0, c0);
      c1 = wmma_f32(a, b1, c1);
    }

    wait_async0();
    __syncthreads();
  }

  // ---------------- transition: encoded tile -> LDS (A-frag layout) ---------
  // Region map after final encode barrier (all disjoint; Wd1 extends past old buffer 0 into buffer 1):
  //   Es  = smem[0      .. 2304)
  //   Wd0 = smem[2304   .. 4608)
  //   Wd1 = smem[4608   .. 6912)
  float* Es = smem;
  {
    float bias0 = eb[t * 32 + l16];
    float bias1 = eb[t * 32 + 16 + l16];
    #pragma unroll
    for (int r = 0; r < 8; ++r) {
      int row = w * 16 + r + 8 * halfSel;     // C layout: lanes>=16 hold M=8..15
      Es[row * ES_STRIDE + l16]      = c0[r] + bias0;
      Es[row * ES_STRIDE + 16 + l16] = c1[r] + bias1;
    }
  }

  // Issue async stage of decoder-weight chunk d (pp in [64d,64d+64)).
  auto stage_dec = [&](int d, int bsel) {
    float* wd = smem + ES_FLOATS + bsel * ES_FLOATS;
    #pragma unroll
    for (int it = 0; it < 4; ++it) {
      int q  = tid + 128 * it;     // 0..511 float4s
      int jj = q & 7;
      int pr = q >> 3;             // 0..63
      const float* src = dw_base + (d * 64 + pr) * 32 + jj * 4;
      async_copy16(src, wd + pr * ES_STRIDE + jj * 4);
    }
  };

  // ---------------- decode: 64 N-tiles of 16x16, K=32, double-buffered ------
  const float* Erow = Es + (w * 16 + l16) * ES_STRIDE;

  stage_dec(0, 0);
  wait_async0();
  __syncthreads();

  for (int d = 0; d < 16; ++d) {
    const int cur = d & 1;
    if (d + 1 < 16) stage_dec(d + 1, cur ^ 1);

    const float* wd = smem + ES_FLOATS + cur * ES_FLOATS;
    #pragma unroll
    for (int s = 0; s < 4; ++s) {
      v8f acc = {};
      const float* Wrow = wd + (s * 16 + l16) * ES_STRIDE;
      #pragma unroll
      for (int k2 = 0; k2 < 32; k2 += 4) {
        int kA = k2 + 2 * halfSel;
        v2f a; a[0] = Erow[kA]; a[1] = Erow[kA + 1];
        int kB = k2 + halfSel;
        v2f b; b[0] = Wrow[kB]; b[1] = Wrow[kB + 2];
        acc = wmma_f32(a, b, acc);
      }
      int pp0 = d * 64 + s * 16;
      float bb = db[t * 1024 + pp0 + l16];
      int i = pp0 >> 5;                 // constant per tile
      int j = (pp0 & 31) + l16;
      size_t outoff = (size_t)t_off + i * 128 + j;
      #pragma unroll
      for (int r = 0; r < 8; ++r) {
        int row = B0 + w * 16 + r + 8 * halfSel;
        float v = acc[r] + bb;
        out[(size_t)row * HW_TOTAL + outoff] = 1.0f / (1.0f + __expf(-v));
      }
    }

    wait_async0();
    __syncthreads();
  }
}

extern "C" void kernel_launch(void* const* d_in, const int* in_sizes, int n_in,
                              void* d_out, int out_size, void* d_ws, size_t ws_size,
                              hipStream_t stream) {
  const float* x  = (const float*)d_in[0];
  const float* ew = (const float*)d_in[1];
  const float* eb = (const float*)d_in[2];
  const float* dw = (const float*)d_in[3];
  const float* db = (const float*)d_in[4];
  float* out = (float*)d_out;

  int Brows = in_sizes[0] / HW_TOTAL;          // 2048
  int blocks = (Brows / 64) * 32;              // 32 row-groups * 32 patches
  lca_fused_kernel<<<blocks, 128, 0, stream>>>(x, ew, eb, dw, db, out, Brows);
}